// ShadingLayer_71193377899359
// MI455X (gfx1250) — compile-verified
//
#include <hip/hip_runtime.h>

// SH constants (fp32, precomputed from the reference):
//   C1 = pi * 1/sqrt(4pi), C2 = (2pi/3)*sqrt(3/(4pi)), C3 = (pi/4)*0.5*sqrt(5/(4pi)),
//   C4 = (pi/4)*3*sqrt(5/(12pi)), C5 = (pi/4)*3*sqrt(5/(48pi))
#define C1f 0.8862269254527580f
#define C2f 1.0233267079464885f
#define C3f 0.2477079561224014f
#define C4f 0.8580855308097834f
#define C5f 0.4290427654048917f

#define NPIX (512 * 512)         // pixels per (batch, channel) plane
#define GRID_X 256               // tile-groups per batch
#define WAVES_PER_BLOCK 8        // 256 threads, wave32
#define TILES_PER_WAVE 8         // 256 blocks * 8 waves * 8 tiles * 16 px = 262144

typedef float v2f __attribute__((ext_vector_type(2)));
typedef float v8f __attribute__((ext_vector_type(8)));

__global__ __launch_bounds__(256)
void ShadingLayer_71193377899359_kernel(const float* __restrict__ normals,   // [B,3,512,512]
                                        const float* __restrict__ fc_light,  // [B,27]
                                        float* __restrict__ out)             // [B,3,512,512]
{
    const int lane = threadIdx.x & 31;
    const int wave = threadIdx.x >> 5;
    const int m    = lane & 15;   // A-row (channel) / B-col (pixel-in-tile)
    const int half = lane >> 4;   // selects which K pair this lane supplies
    const int b    = blockIdx.y;

    // ---- A matrix = L (16x12, rows = channels; only rows 0..2 nonzero) ----
    // f32 A-layout (ISA 7.12.2): lane m holds K = half*2, half*2+1 in VGPR0/1.
    const float* L = fc_light + b * 27;
    v2f A0 = {0.f, 0.f}, A1 = {0.f, 0.f}, A2 = {0.f, 0.f};
    if (m < 3) {
        const int k0 = half * 2;          // chunk 0: K = 0..3  (H1..H4)
        A0.x = L[m * 9 + k0];
        A0.y = L[m * 9 + k0 + 1];
        const int k1 = 4 + half * 2;      // chunk 1: K = 4..7  (H5..H8)
        A1.x = L[m * 9 + k1];
        A1.y = L[m * 9 + k1 + 1];
        if (half == 0) A2.x = L[m * 9 + 8];   // chunk 2: K = 8 (H9), K9..11 = 0
    }

    const float* nx_p = normals + (size_t)b * 3 * NPIX;
    const float* ny_p = nx_p + NPIX;
    const float* nz_p = nx_p + 2 * NPIX;
    float* outb = out + (size_t)b * 3 * NPIX;

    const int tile0 = blockIdx.x * (WAVES_PER_BLOCK * TILES_PER_WAVE) + wave * TILES_PER_WAVE;

    for (int t = 0; t < TILES_PER_WAVE; ++t) {
        const int p = (tile0 + t) * 16 + m;   // both lane-halves handle the same 16 pixels

        // prefetch next tile into GL2 (global_prefetch_b8; speculative, drop-on-fault)
        __builtin_prefetch(&nx_p[p + 16], 0, 0);
        __builtin_prefetch(&ny_p[p + 16], 0, 0);
        __builtin_prefetch(&nz_p[p + 16], 0, 0);

        // Regular-temporal loads: input is the stream we WANT resident in the
        // 192MB L2 (201MB input nearly fits; reused across timed graph replays).
        const float nx = nx_p[p];
        const float ny = ny_p[p];
        const float nz = nz_p[p];

        const float H1 = C1f;
        const float H2 = C2f * nz;
        const float H3 = C2f * nx;
        const float H4 = C2f * ny;
        const float H5 = C3f * (2.f * nz * nz - nx * nx - ny * ny);
        const float H6 = C4f * nx * nz;
        const float H7 = C4f * ny * nz;
        const float H8 = C5f * (nx * nx - ny * ny);
        const float H9 = C4f * nx * ny;

        // ---- B matrix = basis (K x 16): f32 B-layout mirrors A:
        // lane n supplies rows K = half*2 (VGPR0) and half*2+1 (VGPR1) for pixel n.
        v2f B0, B1, B2;
        B0.x = half ? H3 : H1;   B0.y = half ? H4 : H2;   // K0..K3
        B1.x = half ? H7 : H5;   B1.y = half ? H8 : H6;   // K4..K7
        B2.x = half ? 0.f : H9;  B2.y = 0.f;              // K8, rest zero

        // D = A*B (+0): 3 chained v_wmma_f32_16x16x4_f32 cover K=0..11.
        // EXEC is all-1s here (no divergence at wave scope).
        v8f Cc = {};
        Cc = __builtin_amdgcn_wmma_f32_16x16x4_f32(false, A0, false, B0, (short)0, Cc, false, false);
        Cc = __builtin_amdgcn_wmma_f32_16x16x4_f32(false, A1, false, B1, (short)0, Cc, false, false);
        Cc = __builtin_amdgcn_wmma_f32_16x16x4_f32(false, A2, false, B2, (short)0, Cc, false, false);

        // D layout: VGPR j, lanes 0..15 = row M=j = channel j, col = pixel n.
        // Lanes 16..31 hold rows 8..10 (garbage) -> masked off.
        // Output is write-once/never-read: non-temporal stores keep it from
        // evicting the input stream out of L2.
        if (half == 0) {
            __builtin_nontemporal_store(Cc[0], &outb[p]);
            __builtin_nontemporal_store(Cc[1], &outb[NPIX + p]);
            __builtin_nontemporal_store(Cc[2], &outb[2 * NPIX + p]);
        }
    }
}

extern "C" void kernel_launch(void* const* d_in, const int* in_sizes, int n_in,
                              void* d_out, int out_size, void* d_ws, size_t ws_size,
                              hipStream_t stream) {
    const float* normals  = (const float*)d_in[0];   // [B,3,512,512] fp32
    const float* fc_light = (const float*)d_in[1];   // [B,27] fp32
    float* out = (float*)d_out;

    const int B = in_sizes[1] / 27;                  // 64
    dim3 grid(GRID_X, B);
    dim3 block(WAVES_PER_BLOCK * 32);
    ShadingLayer_71193377899359_kernel<<<grid, block, 0, stream>>>(normals, fc_light, out);
}